// SimpleLightGCN_75531294867819
// MI455X (gfx1250) — compile-verified
//
#include <hip/hip_runtime.h>
#include <stdint.h>

#define NUM_USERS 100000
#define NUM_ITEMS 50000
#define EMB_DIM   64
#define NUM_EDGES 1000000
#define NTOT      (NUM_USERS + NUM_ITEMS)

// ---------------------------------------------------------------------------
// elementwise helpers (float4-wide)
// ---------------------------------------------------------------------------
__global__ __launch_bounds__(256) void lgcn_zero4(float4* __restrict__ p, long n4) {
  long i = (long)blockIdx.x * blockDim.x + threadIdx.x;
  if (i < n4) p[i] = make_float4(0.f, 0.f, 0.f, 0.f);
}

// acc = 0.25*emb  AND  x0 = 0   (same N*D/4 index space; saves a full pass)
__global__ __launch_bounds__(256) void lgcn_init_acc_zero4(const float4* __restrict__ emb,
                                                           float4* __restrict__ acc,
                                                           float4* __restrict__ x0,
                                                           long n4) {
  long i = (long)blockIdx.x * blockDim.x + threadIdx.x;
  if (i < n4) {
    float4 e = emb[i];
    acc[i] = make_float4(0.25f * e.x, 0.25f * e.y, 0.25f * e.z, 0.25f * e.w);
    x0[i]  = make_float4(0.f, 0.f, 0.f, 0.f);
  }
}

// acc += 0.25*x  AND  zero the other ping-pong buffer for the next layer
__global__ __launch_bounds__(256) void lgcn_axpy_zero4(const float4* __restrict__ x,
                                                       float4* __restrict__ acc,
                                                       float4* __restrict__ zbuf,
                                                       long n4) {
  long i = (long)blockIdx.x * blockDim.x + threadIdx.x;
  if (i < n4) {
    float4 a = acc[i];
    float4 b = x[i];
    a.x += 0.25f * b.x; a.y += 0.25f * b.y;
    a.z += 0.25f * b.z; a.w += 0.25f * b.w;
    acc[i]  = a;
    zbuf[i] = make_float4(0.f, 0.f, 0.f, 0.f);
  }
}

// final layer: acc += 0.25*x (no buffer to recycle)
__global__ __launch_bounds__(256) void lgcn_axpy4(const float4* __restrict__ x,
                                                  float4* __restrict__ acc, long n4) {
  long i = (long)blockIdx.x * blockDim.x + threadIdx.x;
  if (i < n4) {
    float4 a = acc[i];
    float4 b = x[i];
    a.x += 0.25f * b.x; a.y += 0.25f * b.y;
    a.z += 0.25f * b.z; a.w += 0.25f * b.w;
    acc[i] = a;
  }
}

__device__ __forceinline__ void atomic_add_f32(float* p, float v) {
  __hip_atomic_fetch_add(p, v, __ATOMIC_RELAXED, __HIP_MEMORY_SCOPE_AGENT);
}

// ---------------------------------------------------------------------------
// degree count (symmetric: each undirected edge adds 1 to both endpoints)
// ---------------------------------------------------------------------------
__global__ __launch_bounds__(256) void lgcn_degree(const int* __restrict__ ei,
                                                   float* __restrict__ deg) {
  long t = (long)blockIdx.x * blockDim.x + threadIdx.x;
  if (t >= NUM_EDGES) return;
  // stream-prefetch the edge list ahead of this wavefront (global_prefetch_b8)
  if (t + 16384 < NUM_EDGES) {
    __builtin_prefetch(ei + t + 16384, 0, 0);
    __builtin_prefetch(ei + NUM_EDGES + t + 16384, 0, 0);
  }
  int u = ei[t];                         // user id
  int v = ei[NUM_EDGES + t] + NUM_USERS; // item id offset into shared space
  atomic_add_f32(&deg[u], 1.0f);
  atomic_add_f32(&deg[v], 1.0f);
}

// in-place deg -> dinv = deg>0 ? rsqrt(deg) : 0
__global__ __launch_bounds__(256) void lgcn_dinv(float* __restrict__ p, int n) {
  int i = blockIdx.x * blockDim.x + threadIdx.x;
  if (i < n) {
    float d = p[i];
    p[i] = (d > 0.f) ? __frsqrt_rn(d) : 0.f;
  }
}

// ---------------------------------------------------------------------------
// edge-parallel LGConv scatter: one block = 16 edges; 16 lanes per edge,
// each lane handles one float4 chunk of the 64-wide embedding row.
// Edge tile (u-row + v-row) is staged into LDS with gfx1250 async-to-LDS
// loads; per-edge GCN norms are computed once into LDS by wave 0.
// ---------------------------------------------------------------------------
__global__ __launch_bounds__(256) void lgcn_scatter(const int* __restrict__ ei,
                                                    const float* __restrict__ dinv,
                                                    const float* __restrict__ xcur,
                                                    float* __restrict__ xnext) {
  __shared__ int   s_edge[32];   // [0..15] = u ids, [16..31] = raw item ids
  __shared__ float s_norm[16];   // per-edge symmetric normalization

  const int  tid = threadIdx.x;
  const long e0  = (long)blockIdx.x * 16;

  if (tid < 32) {
    const int lane = tid;
    // byte offset into edge_index: users row at [0,E), items row at [E,2E)
    unsigned voff = (unsigned)((lane < 16
                                    ? (e0 + lane)
                                    : ((long)NUM_EDGES + e0 + (lane - 16))) * 4);
    // LDS destination is a single 32-bit VGPR: low 32 bits of the flat
    // shared-space address are the LDS byte offset.
    unsigned lds_addr = (unsigned)(uintptr_t)&s_edge[lane];
    // gfx1250 async DMA: global -> LDS, one dword per lane, tracked by ASYNCcnt
    asm volatile("global_load_async_to_lds_b32 %0, %1, %2"
                 :: "v"(lds_addr), "v"(voff), "s"(ei)
                 : "memory");
    asm volatile("s_wait_asynccnt 0" ::: "memory");
    if (lane < 16) {
      int u = s_edge[lane];
      int v = s_edge[16 + lane] + NUM_USERS;
      s_norm[lane] = dinv[u] * dinv[v];
    }
  }
  __syncthreads();

  const int el = tid >> 4;   // edge within tile: 0..15
  const int ch = tid & 15;   // float4 chunk:     0..15
  const int u  = s_edge[el];
  const int v  = s_edge[16 + el] + NUM_USERS;
  const float nrm = s_norm[el];

  const float4 xu = ((const float4*)xcur)[(long)u * 16 + ch];
  const float4 xv = ((const float4*)xcur)[(long)v * 16 + ch];

  float* pv = xnext + (long)v * EMB_DIM + ch * 4;
  float* pu = xnext + (long)u * EMB_DIM + ch * 4;
  // message u -> v and reverse message v -> u (undirected graph)
  atomic_add_f32(pv + 0, nrm * xu.x);
  atomic_add_f32(pv + 1, nrm * xu.y);
  atomic_add_f32(pv + 2, nrm * xu.z);
  atomic_add_f32(pv + 3, nrm * xu.w);
  atomic_add_f32(pu + 0, nrm * xv.x);
  atomic_add_f32(pu + 1, nrm * xv.y);
  atomic_add_f32(pu + 2, nrm * xv.z);
  atomic_add_f32(pu + 3, nrm * xv.w);
}

// ---------------------------------------------------------------------------
// host-side orchestration (graph-capture safe: launches only)
// ---------------------------------------------------------------------------
extern "C" void kernel_launch(void* const* d_in, const int* in_sizes, int n_in,
                              void* d_out, int out_size, void* d_ws, size_t ws_size,
                              hipStream_t stream) {
  (void)in_sizes; (void)n_in; (void)out_size; (void)ws_size;

  const int*   ei  = (const int*)d_in[0];    // (2, E) flattened: users then items
  const float* emb = (const float*)d_in[1];  // (N, 64) fp32
  float*       acc = (float*)d_out;          // (N, 64) fp32 output

  // workspace layout (floats): dinv[N] | xA[N*64] | xB[N*64]  (~77.4 MB)
  float* dinv = (float*)d_ws;
  float* xA   = dinv + NTOT;                       // 150000*4 B keeps 16B alignment
  float* xB   = xA + (size_t)NTOT * EMB_DIM;

  const long nd4  = (long)NTOT * EMB_DIM / 4;      // 2,400,000
  const int  gN   = (NTOT + 255) / 256;
  const int  gND4 = (int)((nd4 + 255) / 256);
  const int  gE   = (NUM_EDGES + 255) / 256;
  const int  gS   = NUM_EDGES / 16;                // 62500, exact

  // degrees -> dinv (in place)
  lgcn_zero4<<<(NTOT / 4 + 255) / 256, 256, 0, stream>>>((float4*)dinv, NTOT / 4);
  lgcn_degree<<<gE, 256, 0, stream>>>(ei, dinv);
  lgcn_dinv<<<gN, 256, 0, stream>>>(dinv, NTOT);

  // acc = emb/4, xA = 0 (fused single pass)
  lgcn_init_acc_zero4<<<gND4, 256, 0, stream>>>((const float4*)emb, (float4*)acc,
                                                (float4*)xA, nd4);

  // layer 1: emb -> xA; accumulate and zero xB for layer 2
  lgcn_scatter<<<gS, 256, 0, stream>>>(ei, dinv, emb, xA);
  lgcn_axpy_zero4<<<gND4, 256, 0, stream>>>((const float4*)xA, (float4*)acc,
                                            (float4*)xB, nd4);

  // layer 2: xA -> xB; accumulate and zero xA for layer 3
  // (legal: layer-2 scatter already consumed xA before this axpy runs)
  lgcn_scatter<<<gS, 256, 0, stream>>>(ei, dinv, xA, xB);
  lgcn_axpy_zero4<<<gND4, 256, 0, stream>>>((const float4*)xB, (float4*)acc,
                                            (float4*)xA, nd4);

  // layer 3: xB -> xA; final accumulate
  lgcn_scatter<<<gS, 256, 0, stream>>>(ei, dinv, xB, xA);
  lgcn_axpy4<<<gND4, 256, 0, stream>>>((const float4*)xA, (float4*)acc, nd4);
}